// GPWorldModel_64879775973531
// MI455X (gfx1250) — compile-verified
//
#include <hip/hip_runtime.h>
#include <hip/hip_bf16.h>
#include <math.h>

// Problem dims (fixed by the reference)
#define Bb 2
#define Ee 16
#define Uu 8
#define Nn 400
#define Dd 24
#define DP 25   // padded LDS row stride (gcd(25,64)=1 -> conflict-free)
#define NT (Nn / 16)   // 25 tiles per dimension

// Workspace layout (float offsets)
#define WS_M     0        // B*D                 = 48
#define WS_S     48       // B*D*D               = 1152
#define WS_Z     1200     // B*E*N*D             = 307200
#define WS_LOGK  308400   // B*E*N               = 12800
#define WS_PRED  321200   // B*E                 = 32   (pred_mean, already *c)
#define WS_V     321232   // B*D*E               = 768  (V, already *c)
#define WS_SCALE 322000   // B*E*E               = 512
#define WS_SBLB  322512   // B*E*E               = 512
#define WS_TR    323024   // B*E                 = 32
#define WS_TOTAL 323056

typedef __attribute__((ext_vector_type(2))) float v2f;
typedef __attribute__((ext_vector_type(8))) float v8f;

#if defined(__gfx1250__) && __has_builtin(__builtin_amdgcn_wmma_f32_16x16x4_f32)
#define USE_WMMA_F32 1
#else
#define USE_WMMA_F32 0
#endif

// ---------- shared helpers: 24x24 SPD Cholesky (in LDS, row stride DP) ----------
__device__ __forceinline__ void chol24(float* A, int tid) {
  for (int k = 0; k < Dd; ++k) {
    __syncthreads();
    if (tid == 0) A[k * DP + k] = sqrtf(A[k * DP + k]);
    __syncthreads();
    if (tid > k && tid < Dd) A[tid * DP + k] /= A[k * DP + k];
    __syncthreads();
    if (tid > k && tid < Dd) {
      float lik = A[tid * DP + k];
      for (int j = k + 1; j <= tid; ++j) A[tid * DP + j] -= lik * A[j * DP + k];
    }
  }
  __syncthreads();
}

// Ainv = (L L^T)^{-1}; one thread per column (tid<24). L is lower triangle of A.
__device__ __forceinline__ void inv_from_chol24(const float* L, float* Ainv, int tid) {
  if (tid < Dd) {
    const int j = tid;
    float y[Dd];
#pragma unroll
    for (int i = 0; i < Dd; ++i) {
      float s = (i == j) ? 1.0f : 0.0f;
      for (int k = 0; k < i; ++k) s -= L[i * DP + k] * y[k];
      y[i] = s / L[i * DP + i];
    }
#pragma unroll
    for (int i = Dd - 1; i >= 0; --i) {
      float s = y[i];
      for (int k = i + 1; k < Dd; ++k) s -= L[k * DP + i] * y[k];
      y[i] = s / L[i * DP + i];
    }
#pragma unroll
    for (int i = 0; i < Dd; ++i) Ainv[i * DP + j] = y[i];
  }
  __syncthreads();
}

// ---------------- Kernel A: m = [m_x,m_u]; S = [[s_x, s_x@c_xu],[.T, s_u]] ----------------
__global__ __launch_bounds__(256) void ka_build_mS(const float* __restrict__ m_x,
                                                   const float* __restrict__ s_x,
                                                   const float* __restrict__ m_u,
                                                   const float* __restrict__ s_u,
                                                   const float* __restrict__ c_xu,
                                                   float* __restrict__ ws_m,
                                                   float* __restrict__ ws_S) {
  const int b = blockIdx.x;
  const int tid = threadIdx.x;
  __shared__ float sc[Ee * Uu];
  if (tid < Ee * Uu) {
    const int i = tid / Uu, j = tid % Uu;
    float s = 0.f;
#pragma unroll
    for (int k = 0; k < Ee; ++k) s += s_x[b * Ee * Ee + i * Ee + k] * c_xu[b * Ee * Uu + k * Uu + j];
    sc[i * Uu + j] = s;
  }
  if (tid < Dd) ws_m[b * Dd + tid] = (tid < Ee) ? m_x[b * Ee + tid] : m_u[b * Uu + (tid - Ee)];
  __syncthreads();
  for (int idx = tid; idx < Dd * Dd; idx += 256) {
    const int d = idx / Dd, e = idx % Dd;
    float v;
    if (d < Ee && e < Ee)      v = s_x[b * Ee * Ee + d * Ee + e];
    else if (d < Ee)           v = sc[d * Uu + (e - Ee)];
    else if (e < Ee)           v = sc[e * Uu + (d - Ee)];
    else                       v = s_u[b * Uu * Uu + (d - Ee) * Uu + (e - Ee)];
    ws_S[b * Dd * Dd + idx] = v;
  }
}

// ---------------- Kernel B: per (b,e): Bmat solve, lb, pred_mean, V, z, logk ----------------
__global__ __launch_bounds__(256) void kb_predict(const float* __restrict__ X_train,
                                                  const float* __restrict__ lengthscales,
                                                  const float* __restrict__ variances,
                                                  const float* __restrict__ beta,
                                                  const float* __restrict__ ws_m,
                                                  const float* __restrict__ ws_S,
                                                  float* __restrict__ ws_z,
                                                  float* __restrict__ ws_logk,
                                                  float* __restrict__ ws_pred,
                                                  float* __restrict__ ws_V) {
  const int e = blockIdx.x;
  const int b = blockIdx.y;
  const int tid = threadIdx.x;

  __shared__ float A[Dd * DP];
  __shared__ float Binv[Dd * DP];
  __shared__ float red[256 * DP];
  __shared__ float mS[Dd], iLs[Dd], iLq[Dd];
  __shared__ float cS[1];

  if (tid < Dd) {
    const float il = 1.0f / lengthscales[e * Dd + tid];
    iLs[tid] = il;
    iLq[tid] = il * il;
    mS[tid] = ws_m[b * Dd + tid];
  }
  __syncthreads();
  for (int idx = tid; idx < Dd * Dd; idx += 256) {
    const int d = idx / Dd, e2 = idx % Dd;
    A[d * DP + e2] = iLs[d] * ws_S[b * Dd * Dd + idx] * iLs[e2] + ((d == e2) ? 1.0f : 0.0f);
  }
  chol24(A, tid);
  if (tid == 0) {
    float s = 0.f;
#pragma unroll
    for (int k = 0; k < Dd; ++k) s += logf(A[k * DP + k]);
    // c = var_e * exp(-0.5 * logdetB), logdetB = 2*s
    cS[0] = variances[e] * __expf(-s);
  }
  inv_from_chol24(A, Binv, tid);  // ends with barrier (also covers cS)

  const float lvar = logf(variances[e]);
  float vacc[Dd];
#pragma unroll
  for (int d = 0; d < Dd; ++d) vacc[d] = 0.f;
  float predacc = 0.f;

  for (int n = tid; n < Nn; n += 256) {
    float inp[Dd], invN[Dd];
#pragma unroll
    for (int d = 0; d < Dd; ++d) {
      const float x = X_train[n * Dd + d] - mS[d];
      inp[d] = x;
      invN[d] = x * iLs[d];
      ws_z[((size_t)(b * Ee + e) * Nn + n) * Dd + d] = x * iLq[d];
    }
    float q = 0.f, sk = 0.f;
    float td[Dd];
#pragma unroll
    for (int d = 0; d < Dd; ++d) {
      float s = 0.f;
#pragma unroll
      for (int k = 0; k < Dd; ++k) s += invN[k] * Binv[k * DP + d];
      td[d] = s;
      q += s * invN[d];
      sk += inp[d] * inp[d] * iLq[d];
    }
    const float lb = __expf(-0.5f * q) * beta[e * Nn + n];
    predacc += lb;
#pragma unroll
    for (int d = 0; d < Dd; ++d) vacc[d] += td[d] * iLs[d] * lb;
    ws_logk[(b * Ee + e) * Nn + n] = lvar - 0.5f * sk;
  }

  // deterministic tree reduction of [V(24), pred(1)]
#pragma unroll
  for (int d = 0; d < Dd; ++d) red[tid * DP + d] = vacc[d];
  red[tid * DP + Dd] = predacc;
  __syncthreads();
  for (int s2 = 128; s2 > 0; s2 >>= 1) {
    if (tid < s2) {
#pragma unroll
      for (int d = 0; d <= Dd; ++d) red[tid * DP + d] += red[(tid + s2) * DP + d];
    }
    __syncthreads();
  }
  const float cbe = cS[0];
  if (tid < Dd) ws_V[(b * Dd + tid) * Ee + e] = red[tid] * cbe;  // V[b,d,e] (pre-scaled by c)
  if (tid == 0) ws_pred[b * Ee + e] = red[Dd] * cbe;             // pred_mean (pre-scaled)
}

// ---------------- Kernel C: per (b,a,c): Q2, scale, WMMA cross + fused exp reductions ----------------
// dynamic LDS layout (floats)
#define C_ZC   0
#define C_PS   (C_ZC + Nn * DP)        // 10000
#define C_ZA   (C_PS + Nn * DP)        // 20000
#define C_A    (C_ZA + Nn * DP)        // 30000
#define C_MI   (C_A + Dd * DP)         // 30600
#define C_Q2   (C_MI + Dd * DP)        // 31200
#define C_U    (C_Q2 + Dd * DP)        // 31800
#define C_V    (C_U + Nn)              // 32200
#define C_WA   (C_V + Nn)              // 32600
#define C_WC   (C_WA + Nn)             // 33000
#define C_SL   (C_WC + Nn)             // 33400
#define C_REDA (C_SL + Dd)             // 33424
#define C_REDT (C_REDA + 256)          // 33680
#define C_TOT  (C_REDT + 256)          // 33936 floats = 135744 bytes

// one 16x16 G-tile: 6x V_WMMA_F32_16X16X4_F32 with row-stationary A fragments
__device__ __forceinline__ void gtile(const v2f* af, const float* zc, int j0, int m, int hl,
                                      float* accv) {
#if USE_WMMA_F32
  v8f acc = {0.f, 0.f, 0.f, 0.f, 0.f, 0.f, 0.f, 0.f};
#pragma unroll
  for (int kk = 0; kk < 6; ++kk) {
    // B frag 4x16 of z_c^T: B[k][n] = zc[j0+n][4*kk+k]; lanes 0-15 K={0,1}, 16-31 K={2,3}
    v2f bv;
    bv[0] = zc[(j0 + m) * DP + 4 * kk + 2 * hl + 0];
    bv[1] = zc[(j0 + m) * DP + 4 * kk + 2 * hl + 1];
    acc = __builtin_amdgcn_wmma_f32_16x16x4_f32(false, af[kk], false, bv,
                                                (short)0, acc, false, false);
  }
#pragma unroll
  for (int r = 0; r < 8; ++r) accv[r] = acc[r];
#else
  // scalar fallback matching the C/D lane layout (af[kk] holds P[i][4kk+2hl+{0,1}])
#pragma unroll
  for (int r = 0; r < 8; ++r) accv[r] = 0.f;
  (void)af;
#pragma unroll
  for (int r = 0; r < 8; ++r) {
    // cannot reconstruct full K from fragments in fallback; recompute handled by caller
    (void)zc; (void)j0; (void)m; (void)hl;
  }
#endif
}

__global__ __launch_bounds__(256) void kc_cross(const float* __restrict__ lengthscales,
                                                const float* __restrict__ beta,
                                                const float* __restrict__ invK,
                                                const float* __restrict__ ws_S,
                                                const float* __restrict__ ws_z,
                                                const float* __restrict__ ws_logk,
                                                float* __restrict__ ws_SbLb,
                                                float* __restrict__ ws_tr,
                                                float* __restrict__ ws_scale) {
  extern __shared__ float sm[];
  float* zc = sm + C_ZC;
  float* Ps = sm + C_PS;
  float* za = sm + C_ZA;
  float* A  = sm + C_A;
  float* Mi = sm + C_MI;
  float* Q2 = sm + C_Q2;
  float* u  = sm + C_U;
  float* vv = sm + C_V;
  float* wa = sm + C_WA;
  float* wc = sm + C_WC;
  float* sl = sm + C_SL;
  float* redA = sm + C_REDA;
  float* redT = sm + C_REDT;

  const int c = blockIdx.x, a = blockIdx.y, b = blockIdx.z;
  const int tid = threadIdx.x;

  // lam_d = iLsq_a[d] + iLsq_c[d];  M = I + sqrt(lam) S sqrt(lam)  (SPD)
  if (tid < Dd) {
    const float la = lengthscales[a * Dd + tid];
    const float lc = lengthscales[c * Dd + tid];
    sl[tid] = sqrtf(1.0f / (la * la) + 1.0f / (lc * lc));
  }
  __syncthreads();
  for (int idx = tid; idx < Dd * Dd; idx += 256) {
    const int d = idx / Dd, e = idx % Dd;
    A[d * DP + e] = sl[d] * ws_S[b * Dd * Dd + idx] * sl[e] + ((d == e) ? 1.0f : 0.0f);
  }
  chol24(A, tid);
  if (tid == 0) {
    float s = 0.f;
#pragma unroll
    for (int k = 0; k < Dd; ++k) s += logf(A[k * DP + k]);
    ws_scale[(b * Ee + a) * Ee + c] = __expf(-s);  // exp(-0.5*logdetR), logdetR = 2*s
  }
  inv_from_chol24(A, Mi, tid);
  // Q2 = 0.5 * Lam^-1/2 (I - M^-1) Lam^-1/2
  for (int idx = tid; idx < Dd * Dd; idx += 256) {
    const int d = idx / Dd, e = idx % Dd;
    Q2[d * DP + e] = 0.5f * (((d == e) ? 1.0f : 0.0f) - Mi[d * DP + e]) / (sl[d] * sl[e]);
  }
  // stage z_a, z_c, beta rows into LDS
  for (int idx = tid; idx < Nn * Dd; idx += 256) {
    const int r = idx / Dd, d = idx % Dd;
    za[r * DP + d] = ws_z[((size_t)(b * Ee + a) * Nn + r) * Dd + d];
    zc[r * DP + d] = ws_z[((size_t)(b * Ee + c) * Nn + r) * Dd + d];
  }
  for (int i = tid; i < Nn; i += 256) {
    wa[i] = beta[a * Nn + i];
    wc[i] = beta[c * Nn + i];
  }
  __syncthreads();

  // P = z_a @ Q2 ; u_i = logk[b,a,i] + z_a_i Q2 z_a_i
  for (int i = tid; i < Nn; i += 256) {
    float xs = 0.f;
#pragma unroll
    for (int d = 0; d < Dd; ++d) {
      float s = 0.f;
#pragma unroll
      for (int k = 0; k < Dd; ++k) s += za[i * DP + k] * Q2[k * DP + d];
      Ps[i * DP + d] = s;
      xs += s * za[i * DP + d];
    }
    u[i] = ws_logk[(b * Ee + a) * Nn + i] + xs;
  }
  // v_j = logk[b,c,j] + z_c_j Q2 z_c_j
  for (int j = tid; j < Nn; j += 256) {
    float x2 = 0.f;
#pragma unroll
    for (int d = 0; d < Dd; ++d) {
      float s = 0.f;
#pragma unroll
      for (int k = 0; k < Dd; ++k) s += zc[j * DP + k] * Q2[k * DP + d];
      x2 += s * zc[j * DP + d];
    }
    vv[j] = ws_logk[(b * Ee + c) * Nn + j] + x2;
  }
  __syncthreads();

  // main loop: row-stationary A fragments; per tile 6 WMMA + fused exp reductions.
  // diag (a==c) split out so the trace path has no per-element branches.
  const int lane = tid & 31;
  const int wave = tid >> 5;
  const int hl = lane >> 4;   // 0 or 1 (lane half)
  const int m = lane & 15;
  const bool diag = (a == c);
  float sA = 0.f, sT = 0.f;

  for (int ti = wave; ti < NT; ti += 8) {
    const int i0 = ti * 16;
    // A frags 16x4 of P, kept in registers for the whole tile row
    v2f af[6];
#pragma unroll
    for (int kk = 0; kk < 6; ++kk) {
      af[kk][0] = Ps[(i0 + m) * DP + 4 * kk + 2 * hl + 0];
      af[kk][1] = Ps[(i0 + m) * DP + 4 * kk + 2 * hl + 1];
    }
    float ua[8], war[8];
#pragma unroll
    for (int r = 0; r < 8; ++r) {
      ua[r] = u[i0 + r + 8 * hl];       // C/D layout: VGPR r -> M = r + 8*(lane>=16)
      war[r] = wa[i0 + r + 8 * hl];
    }
    if (!diag) {
      for (int tj = 0; tj < NT; ++tj) {
        const int j0 = tj * 16;
        float accv[8];
#if USE_WMMA_F32
        gtile(af, zc, j0, m, hl, accv);
#else
#pragma unroll
        for (int r = 0; r < 8; ++r) {
          float s = 0.f;
#pragma unroll
          for (int k = 0; k < Dd; ++k) s += Ps[(i0 + r + 8 * hl) * DP + k] * zc[(j0 + m) * DP + k];
          accv[r] = s;
        }
#endif
        const float vj = vv[j0 + m];
        const float wcj = wc[j0 + m];
        float s = 0.f;
#pragma unroll
        for (int r = 0; r < 8; ++r) s += war[r] * __expf(ua[r] + vj + 2.0f * accv[r]);
        sA += wcj * s;
      }
    } else {
      const float* kb0 = invK + ((size_t)a * Nn + i0 + 8 * hl) * Nn + m;
      for (int tj = 0; tj < NT; ++tj) {
        const int j0 = tj * 16;
        const float* kb = kb0 + j0;
        // batch the 8 inv_K loads (immediate offsets r*1600B, single wait)
        float kv[8];
#pragma unroll
        for (int r = 0; r < 8; ++r) kv[r] = kb[(size_t)r * Nn];
        if (tj + 1 < NT) __builtin_prefetch(kb + 16, 0, 0);  // global_prefetch_b8
        float accv[8];
#if USE_WMMA_F32
        gtile(af, zc, j0, m, hl, accv);
#else
#pragma unroll
        for (int r = 0; r < 8; ++r) {
          float s = 0.f;
#pragma unroll
          for (int k = 0; k < Dd; ++k) s += Ps[(i0 + r + 8 * hl) * DP + k] * zc[(j0 + m) * DP + k];
          accv[r] = s;
        }
#endif
        const float vj = vv[j0 + m];
        const float wcj = wc[j0 + m];
        float s = 0.f, st = 0.f;
#pragma unroll
        for (int r = 0; r < 8; ++r) {
          const float val = __expf(ua[r] + vj + 2.0f * accv[r]);
          s += war[r] * val;
          st += kv[r] * val;
        }
        sA += wcj * s;
        sT += st;
      }
    }
  }

  redA[tid] = sA;
  redT[tid] = sT;
  __syncthreads();
  for (int s2 = 128; s2 > 0; s2 >>= 1) {
    if (tid < s2) {
      redA[tid] += redA[tid + s2];
      redT[tid] += redT[tid + s2];
    }
    __syncthreads();
  }
  if (tid == 0) {
    ws_SbLb[(b * Ee + a) * Ee + c] = redA[0];
    if (diag) ws_tr[b * Ee + a] = redT[0];
  }
}

// ---------------- Kernel D: final assembly ----------------
__global__ __launch_bounds__(256) void kd_finalize(const float* __restrict__ m_x,
                                                   const float* __restrict__ s_x,
                                                   const float* __restrict__ variances,
                                                   const float* __restrict__ ws_S,
                                                   const float* __restrict__ ws_V,
                                                   const float* __restrict__ ws_pred,
                                                   const float* __restrict__ ws_SbLb,
                                                   const float* __restrict__ ws_tr,
                                                   const float* __restrict__ ws_scale,
                                                   float* __restrict__ out) {
  __shared__ float Cx[Bb * Ee * Ee];
  const int tid = threadIdx.x;
  // Cx[b,i,j] = sum_d S[b, i(<16), d] * V[b, d, j]
  for (int idx = tid; idx < Bb * Ee * Ee; idx += 256) {
    const int b = idx >> 8, i = (idx >> 4) & 15, j = idx & 15;
    float s = 0.f;
#pragma unroll
    for (int d = 0; d < Dd; ++d) s += ws_S[b * Dd * Dd + i * Dd + d] * ws_V[(b * Dd + d) * Ee + j];
    Cx[idx] = s;
  }
  __syncthreads();
  if (tid < Bb * Ee) out[tid] = m_x[tid] + ws_pred[tid];  // next_mean (B,E)
  for (int idx = tid; idx < Bb * Ee * Ee; idx += 256) {
    const int b = idx >> 8, i = (idx >> 4) & 15, j = idx & 15;
    const float dg = (i == j) ? 1.0f : 0.0f;
    const float pmi = ws_pred[b * Ee + i], pmj = ws_pred[b * Ee + j];
    const float sd = (ws_SbLb[idx] - dg * ws_tr[b * Ee + i]) * ws_scale[idx]
                   + dg * variances[i] - pmi * pmj;
    out[Bb * Ee + idx] = s_x[idx] + sd + Cx[idx] + Cx[(b * Ee + j) * Ee + i];
  }
}

extern "C" void kernel_launch(void* const* d_in, const int* in_sizes, int n_in,
                              void* d_out, int out_size, void* d_ws, size_t ws_size,
                              hipStream_t stream) {
  const float* m_x    = (const float*)d_in[0];
  const float* s_x    = (const float*)d_in[1];
  const float* m_u    = (const float*)d_in[2];
  const float* s_u    = (const float*)d_in[3];
  const float* c_xu   = (const float*)d_in[4];
  const float* Xtr    = (const float*)d_in[5];
  const float* ls     = (const float*)d_in[6];
  const float* var    = (const float*)d_in[7];
  // d_in[8] = noises (unused: folded into inv_K/beta upstream)
  const float* invK   = (const float*)d_in[9];
  const float* beta   = (const float*)d_in[10];
  float* out = (float*)d_out;
  float* ws = (float*)d_ws;

  float* ws_m     = ws + WS_M;
  float* ws_S     = ws + WS_S;
  float* ws_z     = ws + WS_Z;
  float* ws_logk  = ws + WS_LOGK;
  float* ws_pred  = ws + WS_PRED;
  float* ws_V     = ws + WS_V;
  float* ws_scale = ws + WS_SCALE;
  float* ws_SbLb  = ws + WS_SBLB;
  float* ws_tr    = ws + WS_TR;

  ka_build_mS<<<dim3(Bb), 256, 0, stream>>>(m_x, s_x, m_u, s_u, c_xu, ws_m, ws_S);
  kb_predict<<<dim3(Ee, Bb), 256, 0, stream>>>(Xtr, ls, var, beta, ws_m, ws_S,
                                               ws_z, ws_logk, ws_pred, ws_V);
  kc_cross<<<dim3(Ee, Ee, Bb), 256, C_TOT * sizeof(float), stream>>>(
      ls, beta, invK, ws_S, ws_z, ws_logk, ws_SbLb, ws_tr, ws_scale);
  kd_finalize<<<dim3(1), 256, 0, stream>>>(m_x, s_x, var, ws_S, ws_V, ws_pred,
                                           ws_SbLb, ws_tr, ws_scale, out);
  (void)in_sizes; (void)n_in; (void)out_size; (void)ws_size;
}